// TorchWrithe_45818711113798
// MI455X (gfx1250) — compile-verified
//
#include <hip/hip_runtime.h>
#include <math.h>
#include <stdint.h>

// Writhe (Klenin–Langowski Gauss integral) over all segment pairs per frame.
// xyz: (B, 300, 3) f32   segments: (S, 4) i32   out: (B, S) f32
//
// VALU-bound elementwise kernel (no matmul structure -> WMMA not applicable).
// CDNA5 path used: async global->LDS staging of each frame's coordinates
// (GLOBAL_LOAD_ASYNC_TO_LDS_B128, tracked by ASYNCcnt + s_wait_asynccnt).

#define N_ATOMS      300
#define FRAME_FLOATS (N_ATOMS * 3)   // 900 floats = 3600 B per frame
#define STAGE_LANES  (FRAME_FLOATS / 4) // 225 lanes x 16 B
#define TPB          256

struct V3 { float x, y, z; };

__device__ __forceinline__ V3 vsub(V3 a, V3 b) { return {a.x - b.x, a.y - b.y, a.z - b.z}; }
__device__ __forceinline__ float vdot(V3 a, V3 b) { return a.x * b.x + a.y * b.y + a.z * b.z; }
__device__ __forceinline__ V3 vcross(V3 a, V3 b) {
    return {a.y * b.z - a.z * b.y,
            a.z * b.x - a.x * b.z,
            a.x * b.y - a.y * b.x};
}
__device__ __forceinline__ V3 vnorm(V3 a) {
    float inv = 1.0f / sqrtf(vdot(a, a));   // matches x / ||x|| of reference
    return {a.x * inv, a.y * inv, a.z * inv};
}
__device__ __forceinline__ float clamp1(float x) {
    return fminf(fmaxf(x, -1.0f), 1.0f);
}

__global__ void __launch_bounds__(TPB)
writhe_kernel(const float* __restrict__ xyz,
              const int*   __restrict__ segs,
              float*       __restrict__ out,
              int S)
{
    __shared__ __align__(16) float sxyz[FRAME_FLOATS];

    const int tid = threadIdx.x;
    const int b   = blockIdx.y;
    const float* fbase = xyz + (size_t)b * FRAME_FLOATS;

    // ---- CDNA5 async global -> LDS staging of this frame (3600 B) ----------
    // GVS addressing: mem = SGPR64 base + VGPR32 offset; VDST VGPR holds the
    // LDS byte offset (low 32 bits of the generic shared pointer).
    if (tid < STAGE_LANES) {
        unsigned lds_off = (unsigned)(uintptr_t)(&sxyz[0]) + (unsigned)(tid * 16);
        unsigned g_off   = (unsigned)(tid * 16);
        asm volatile("global_load_async_to_lds_b128 %0, %1, %2"
                     :
                     : "v"(lds_off), "v"(g_off), "s"(fbase)
                     : "memory");
    }
    asm volatile("s_wait_asynccnt 0x0" ::: "memory");
    __syncthreads();

    // ---- per-(frame, pair) writhe contribution -----------------------------
    const int s = blockIdx.x * TPB + tid;
    if (s >= S) return;

    const int4 q = ((const int4*)segs)[s];   // (start1, end1, start2, end2)

    const float* sp = sxyz;
    V3 p0 = {sp[3 * q.x], sp[3 * q.x + 1], sp[3 * q.x + 2]};
    V3 p1 = {sp[3 * q.y], sp[3 * q.y + 1], sp[3 * q.y + 2]};
    V3 p2 = {sp[3 * q.z], sp[3 * q.z + 1], sp[3 * q.z + 2]};
    V3 p3 = {sp[3 * q.w], sp[3 * q.w + 1], sp[3 * q.w + 2]};

    V3 d0 = vnorm(vsub(p2, p0));
    V3 d1 = vnorm(vsub(p2, p1));
    V3 d2 = vnorm(vsub(p3, p0));
    V3 d3 = vnorm(vsub(p3, p1));

    // crosses = nnorm(cross(d[[0,1,3,2]], d[[1,3,2,0]]))
    V3 c0 = vnorm(vcross(d0, d1));
    V3 c1 = vnorm(vcross(d1, d3));
    V3 c2 = vnorm(vcross(d3, d2));
    V3 c3 = vnorm(vcross(d2, d0));

    float omega = asinf(clamp1(vdot(c0, c1)))
                + asinf(clamp1(vdot(c1, c2)))
                + asinf(clamp1(vdot(c2, c3)))
                + asinf(clamp1(vdot(c3, c0)));

    float sg = vdot(vcross(vsub(p3, p2), vsub(p1, p0)), d0);
    float sgn = (sg > 0.0f) ? 1.0f : ((sg < 0.0f) ? -1.0f : 0.0f);

    out[(size_t)b * S + s] = omega * sgn * 0.15915494309189535f; // 1/(2*pi)
}

extern "C" void kernel_launch(void* const* d_in, const int* in_sizes, int n_in,
                              void* d_out, int out_size, void* d_ws, size_t ws_size,
                              hipStream_t stream)
{
    const float* xyz  = (const float*)d_in[0];
    const int*   segs = (const int*)d_in[1];
    float*       out  = (float*)d_out;

    const int B = in_sizes[0] / FRAME_FLOATS;  // 128
    const int S = in_sizes[1] / 4;             // 44253

    dim3 grid((S + TPB - 1) / TPB, B);
    writhe_kernel<<<grid, TPB, 0, stream>>>(xyz, segs, out, S);
}